// MLPDecoder_Causal_90941637525613
// MI455X (gfx1250) — compile-verified
//
#include <hip/hip_runtime.h>
#include <hip/hip_bf16.h>

typedef float v2f __attribute__((ext_vector_type(2)));
typedef float v8f __attribute__((ext_vector_type(8)));
typedef unsigned int u32x4 __attribute__((ext_vector_type(4)));
typedef int i32x4 __attribute__((ext_vector_type(4)));
typedef int i32x8 __attribute__((ext_vector_type(8)));

namespace {
constexpr int kOutPredElems = 8 * 32 * 255 * 4;  // 261120 floats (pred_all)
constexpr int kRelPerB      = 992 * 2;           // 1984 floats per batch

// LDS layout (float offsets)
constexpr int OFF_X    = 0;      // 32*4      = 128
constexpr int OFF_AGG  = 128;    // 32*64     = 2048
constexpr int OFF_W1   = 2176;   // 2*8*64    = 1024
constexpr int OFF_B1   = 3200;   // 2*64      = 128
constexpr int OFF_W2   = 3328;   // 2*64*64   = 8192  (loaded via TDM)
constexpr int OFF_B2   = 11520;  // 2*64      = 128
// ---- phase 1 (edge MLP) region ----
constexpr int OFF_RELG = 11648;  // 992*2     = 1984
constexpr int OFF_MSG1 = 13632;  // 8 waves * 16*64 = 8192
// ---- phase 2 (node MLP) region, aliases phase-1 region ----
constexpr int OFF_P1   = 11648;  // 32*128    = 4096
constexpr int OFF_P2   = 15744;  // 32*128    = 4096
constexpr int SMEM_FLOATS = 21824;               // 87,296 bytes
}

__device__ __forceinline__ v8f wmma4(v2f a, v2f b, v8f c) {
  // V_WMMA_F32_16X16X4_F32 : D = A(16x4) * B(4x16) + C, fp32 tensor path
  return __builtin_amdgcn_wmma_f32_16x16x4_f32(false, a, false, b, (short)0, c,
                                               false, false);
}

// Issue a TDM DMA of `dwords` 32-bit words from `src` into LDS at byte
// offset `lds_addr`, described as an 8192x1 fp32 tile (1 row).
__device__ __forceinline__ void tdm_load_dwords(const void* src,
                                                unsigned int lds_addr,
                                                unsigned int dwords) {
  const unsigned long long ga = (unsigned long long)(uintptr_t)src;
  u32x4 g0;
  g0[0] = 1u;                                   // count=1, user descriptor
  g0[1] = lds_addr;                             // lds_addr (bytes)
  g0[2] = (unsigned int)ga;                     // global_addr[31:0]
  g0[3] = (unsigned int)(ga >> 32) | (2u << 30);// global_addr[56:32] | type=2
  i32x8 g1;
  g1[0] = (int)(2u << 16);                      // data_size = 4 bytes
  g1[1] = (int)((dwords & 0xFFFFu) << 16);      // tensor_dim0[15:0] -> [63:48]
  g1[2] = (int)((dwords >> 16) | (1u << 16));   // tensor_dim0[31:16] | tensor_dim1=1
  g1[3] = (int)(dwords << 16);                  // tile_dim0 -> [127:112]
  g1[4] = 1;                                    // tile_dim1=1, tile_dim2=0
  g1[5] = (int)dwords;                          // tensor_dim0_stride[31:0]
  g1[6] = 0;                                    // stride msbs / dim1 stride
  g1[7] = 0;
  i32x4 z4 = {0, 0, 0, 0};
#if __has_include(<hip/amd_detail/amd_gfx1250_TDM.h>)
  i32x8 z8 = {0, 0, 0, 0, 0, 0, 0, 0};
  __builtin_amdgcn_tensor_load_to_lds(g0, g1, z4, z4, z8, 0);
#else
  __builtin_amdgcn_tensor_load_to_lds(g0, g1, z4, z4, 0);
#endif
}

__global__ __launch_bounds__(256) void nri_decoder_fused(
    const float* __restrict__ inp,   // (8,32,256,4)
    const float* __restrict__ relg,  // (992,2)
    const float* __restrict__ W1,    // (2,8,64)
    const float* __restrict__ b1,    // (2,64)
    const float* __restrict__ W2,    // (2,64,64)
    const float* __restrict__ b2,    // (2,64)
    const float* __restrict__ Wo1,   // (68,128)
    const float* __restrict__ bo1,   // (128)
    const float* __restrict__ Wo2,   // (128,128)
    const float* __restrict__ bo2,   // (128)
    const float* __restrict__ Wo3,   // (128,4)
    const float* __restrict__ bo3,   // (4)
    float* __restrict__ out) {
  __shared__ float smem[SMEM_FLOATS];

  const int tid  = threadIdx.x;
  const int lane = tid & 31;
  const int wave = tid >> 5;
  const int lrow = lane & 15;  // row (A/C) or column-in-tile (B/C)
  const int lhi  = lane >> 4;  // lane-half selects K offset / +8 rows in C
  const int bt   = blockIdx.x;
  const int b    = bt >> 8;
  const int t    = bt & 255;

  // ---- TDM: async-DMA the big weight block (W2, 32KB) into LDS while the
  //      waves stage the small stuff with vector loads ----
  if (wave == 0) {
    tdm_load_dwords(W2, (unsigned int)(uintptr_t)&smem[OFF_W2], 8192u);
  }

  // ---- stage x, small weights, rel_graph; zero aggregates ----
  for (int i = tid; i < 128; i += 256) {
    const int node = i >> 2, d = i & 3;
    smem[OFF_X + i] = inp[((b * 32 + node) * 256 + t) * 4 + d];  // x[b,t,n,d]
  }
  for (int i = tid; i < 2048; i += 256) smem[OFF_AGG + i] = 0.0f;
  for (int i = tid; i < 1024; i += 256) smem[OFF_W1 + i] = W1[i];
  for (int i = tid; i < 128;  i += 256) smem[OFF_B1 + i] = b1[i];
  for (int i = tid; i < 128;  i += 256) smem[OFF_B2 + i] = b2[i];
  for (int i = tid; i < kRelPerB; i += 256) smem[OFF_RELG + i] = relg[i];

  if (wave == 0) __builtin_amdgcn_s_wait_tensorcnt(0);  // W2 DMA complete
  __syncthreads();

  // ================= Phase 1: edge MLP (992 edges = 62 tiles of 16) =========
  for (int tile = wave; tile < 62; tile += 8) {
    const int e0 = tile * 16;

    int recr[8];
    v2f g[8];
#pragma unroll
    for (int r = 0; r < 8; ++r) {
      const int e = e0 + r + 8 * lhi;        // C-fragment row -> edge id
      recr[r] = e / 31;                      // receiver node (pairs are i-major)
      g[r] = *(const v2f*)&smem[OFF_RELG + e * 2];  // (gate_k0, gate_k1)
    }
    // A-fragment row: edge e0+lrow; sender j cols 0..3, receiver i cols 4..7
    const int ea = e0 + lrow;
    const int si = ea / 31;
    const int jj = ea % 31;
    const int sj = jj + (jj >= si ? 1 : 0);
    const v2f aS = *(const v2f*)&smem[OFF_X + sj * 4 + 2 * lhi];
    const v2f aR = *(const v2f*)&smem[OFF_X + si * 4 + 2 * lhi];

    v8f accM[4];
#pragma unroll
    for (int nt = 0; nt < 4; ++nt)
#pragma unroll
      for (int r = 0; r < 8; ++r) accM[nt][r] = 0.0f;

#pragma unroll
    for (int k = 0; k < 2; ++k) {  // expert / edge-type
      // ---- layer 1: 8 -> 64 ----
#pragma unroll
      for (int nt = 0; nt < 4; ++nt) {
        const int n = nt * 16 + lrow;
        const float bb = smem[OFF_B1 + k * 64 + n];
        v8f c;
#pragma unroll
        for (int r = 0; r < 8; ++r) c[r] = bb;
        v2f bv0 = {smem[OFF_W1 + k * 512 + (0 + 2 * lhi) * 64 + n],
                   smem[OFF_W1 + k * 512 + (1 + 2 * lhi) * 64 + n]};
        c = wmma4(aS, bv0, c);
        v2f bv1 = {smem[OFF_W1 + k * 512 + (4 + 2 * lhi) * 64 + n],
                   smem[OFF_W1 + k * 512 + (5 + 2 * lhi) * 64 + n]};
        c = wmma4(aR, bv1, c);
        // ReLU + stage to per-wave LDS scratch to re-fragment as A for layer 2
#pragma unroll
        for (int r = 0; r < 8; ++r)
          smem[OFF_MSG1 + wave * 1024 + (r + 8 * lhi) * 64 + n] =
              __builtin_fmaxf(c[r], 0.0f);
      }
      // ---- layer 2: 64 -> 64 (16 K-steps) ----
      v8f c2[4];
#pragma unroll
      for (int nt = 0; nt < 4; ++nt) {
        const float bb = smem[OFF_B2 + k * 64 + nt * 16 + lrow];
#pragma unroll
        for (int r = 0; r < 8; ++r) c2[nt][r] = bb;
      }
#pragma unroll
      for (int ks = 0; ks < 16; ++ks) {
        const v2f a =
            *(const v2f*)&smem[OFF_MSG1 + wave * 1024 + lrow * 64 + ks * 4 + 2 * lhi];
#pragma unroll
        for (int nt = 0; nt < 4; ++nt) {
          const int n = nt * 16 + lrow;
          v2f bv = {smem[OFF_W2 + k * 4096 + (ks * 4 + 2 * lhi) * 64 + n],
                    smem[OFF_W2 + k * 4096 + (ks * 4 + 2 * lhi + 1) * 64 + n]};
          c2[nt] = wmma4(a, bv, c2[nt]);
        }
      }
      // ReLU, gate by rel_graph, accumulate over experts
#pragma unroll
      for (int nt = 0; nt < 4; ++nt)
#pragma unroll
        for (int r = 0; r < 8; ++r)
          accM[nt][r] += __builtin_fmaxf(c2[nt][r], 0.0f) *
                         (k == 0 ? g[r].x : g[r].y);
    }
    // scatter-add into per-node aggregates (receiver segments via LDS atomics)
#pragma unroll
    for (int nt = 0; nt < 4; ++nt) {
      const int n = nt * 16 + lrow;
#pragma unroll
      for (int r = 0; r < 8; ++r)
        atomicAdd(&smem[OFF_AGG + recr[r] * 64 + n], accM[nt][r]);
    }
  }
  __syncthreads();

  // ================= Phase 2: node MLP (aug = [x | agg], 68 -> 128) =========
  for (int u = wave; u < 16; u += 8) {
    const int rt = u >> 3;
    const int nt = u & 7;
    const int n  = nt * 16 + lrow;
    const int row = rt * 16 + lrow;
    const float bb = bo1[n];
    v8f c;
#pragma unroll
    for (int r = 0; r < 8; ++r) c[r] = bb;
    {  // K-step 0: x columns 0..3
      const v2f a = *(const v2f*)&smem[OFF_X + row * 4 + 2 * lhi];
      v2f bv = {Wo1[(0 + 2 * lhi) * 128 + n], Wo1[(1 + 2 * lhi) * 128 + n]};
      c = wmma4(a, bv, c);
    }
    for (int ks = 1; ks < 17; ++ks) {  // columns 4..67 from agg
      const int k0 = ks * 4;
      const v2f a = *(const v2f*)&smem[OFF_AGG + row * 64 + (k0 - 4) + 2 * lhi];
      v2f bv = {Wo1[(k0 + 2 * lhi) * 128 + n], Wo1[(k0 + 2 * lhi + 1) * 128 + n]};
      c = wmma4(a, bv, c);
    }
#pragma unroll
    for (int r = 0; r < 8; ++r)
      smem[OFF_P1 + (rt * 16 + r + 8 * lhi) * 128 + n] =
          __builtin_fmaxf(c[r], 0.0f);
  }
  __syncthreads();

  // 128 -> 128
  for (int u = wave; u < 16; u += 8) {
    const int rt = u >> 3;
    const int nt = u & 7;
    const int n  = nt * 16 + lrow;
    const float bb = bo2[n];
    v8f c;
#pragma unroll
    for (int r = 0; r < 8; ++r) c[r] = bb;
    for (int ks = 0; ks < 32; ++ks) {
      const v2f a =
          *(const v2f*)&smem[OFF_P1 + (rt * 16 + lrow) * 128 + ks * 4 + 2 * lhi];
      v2f bv = {Wo2[(ks * 4 + 2 * lhi) * 128 + n],
                Wo2[(ks * 4 + 2 * lhi + 1) * 128 + n]};
      c = wmma4(a, bv, c);
    }
#pragma unroll
    for (int r = 0; r < 8; ++r)
      smem[OFF_P2 + (rt * 16 + r + 8 * lhi) * 128 + n] =
          __builtin_fmaxf(c[r], 0.0f);
  }
  __syncthreads();

  // 128 -> 4 (N padded to 16; cols 0..3 valid), residual add, transposed store
  if (wave < 2) {
    const int rt = wave;
    const int n  = lrow;  // output feature d when < 4
    const float bb = (n < 4) ? bo3[n] : 0.0f;
    v8f c;
#pragma unroll
    for (int r = 0; r < 8; ++r) c[r] = bb;
    for (int ks = 0; ks < 32; ++ks) {
      const v2f a =
          *(const v2f*)&smem[OFF_P2 + (rt * 16 + lrow) * 128 + ks * 4 + 2 * lhi];
      v2f bv;
      bv.x = (n < 4) ? Wo3[(ks * 4 + 2 * lhi) * 4 + n] : 0.0f;
      bv.y = (n < 4) ? Wo3[(ks * 4 + 2 * lhi + 1) * 4 + n] : 0.0f;
      c = wmma4(a, bv, c);
    }
    if (t < 255 && n < 4) {
#pragma unroll
      for (int r = 0; r < 8; ++r) {
        const int node = rt * 16 + r + 8 * lhi;
        out[((b * 32 + node) * 255 + t) * 4 + n] =
            smem[OFF_X + node * 4 + n] + c[r];
      }
    }
  }

  // second output: rel_graph broadcast to (B,E,K); written by the t==0 block
  if (t == 0) {
    for (int i = tid; i < kRelPerB; i += 256)
      out[kOutPredElems + b * kRelPerB + i] = relg[i];
  }
}

extern "C" void kernel_launch(void* const* d_in, const int* in_sizes, int n_in,
                              void* d_out, int out_size, void* d_ws,
                              size_t ws_size, hipStream_t stream) {
  (void)in_sizes; (void)n_in; (void)d_ws; (void)ws_size; (void)out_size;
  const float* inp  = (const float*)d_in[0];   // inputs (8,32,256,4)
  // d_in[1]=rel_rec, d_in[2]=rel_send: one-hot, recovered analytically
  const float* relg = (const float*)d_in[3];   // rel_graph (1,1,992,2)
  const float* W1   = (const float*)d_in[4];
  const float* b1   = (const float*)d_in[5];
  const float* W2   = (const float*)d_in[6];
  const float* b2   = (const float*)d_in[7];
  const float* Wo1  = (const float*)d_in[8];
  const float* bo1  = (const float*)d_in[9];
  const float* Wo2  = (const float*)d_in[10];
  const float* bo2  = (const float*)d_in[11];
  const float* Wo3  = (const float*)d_in[12];
  const float* bo3  = (const float*)d_in[13];
  float* out = (float*)d_out;

  nri_decoder_fused<<<8 * 256, 256, 0, stream>>>(
      inp, relg, W1, b1, W2, b2, Wo1, bo1, Wo2, bo2, Wo3, bo3, out);
}